// NodeProcessorContactModule_87608742903957
// MI455X (gfx1250) — compile-verified
//
#include <hip/hip_runtime.h>
#include <hip/hip_bf16.h>

typedef float v2f __attribute__((ext_vector_type(2)));
typedef float v4f __attribute__((ext_vector_type(4)));
typedef float v8f __attribute__((ext_vector_type(8)));

#define DD 128
#define TILE_M 16
#define WPB 4            // waves per block in MLP kernel
#define HSTRIDE 132      // LDS row stride (floats) to spread banks

// ---------------------------------------------------------------------------
// Scatter: one wave (32 lanes) per edge. Lane l handles 4 consecutive floats.
// Edge data is streamed once -> non-temporal loads so the ~52MB of scatter
// accumulators stay resident in the 192MB L2 (where the f32 atomics resolve).
// ---------------------------------------------------------------------------
__global__ __launch_bounds__(256) void scatter_sum_kernel(
    const float* __restrict__ vals, const int* __restrict__ idx,
    float* __restrict__ sums, float* __restrict__ cnt, int n_edges)
{
    int e = blockIdx.x * 8 + (threadIdx.x >> 5);
    if (e >= n_edges) return;
    int lane = threadIdx.x & 31;
    int r = __builtin_nontemporal_load(idx + e);
    const v4f* src = (const v4f*)(vals + (size_t)e * DD);
    v4f v = __builtin_nontemporal_load(src + lane);
    float* dst = sums + (size_t)r * DD + lane * 4;
    atomicAdd(dst + 0, v.x);
    atomicAdd(dst + 1, v.y);
    atomicAdd(dst + 2, v.z);
    atomicAdd(dst + 3, v.w);
    if (lane == 0) atomicAdd(cnt + r, 1.0f);
}

// ---------------------------------------------------------------------------
// One 128-wide K-section of GEMM1/GEMM2 with fp32 WMMA 16x16x4.
// Fragment layout (ISA 7.12.2):
//   A (16x4): lane m=lane&15 holds row m; VGPR v holds K = 2*(lane>>4)+v
//   B (4x16): lane n=lane&15 holds col n; VGPR v holds K = 2*(lane>>4)+v
//   C/D (16x16): VGPR r, lanes 0-15 -> (M=r, N=lane), lanes 16-31 -> (M=r+8)
// ---------------------------------------------------------------------------
__device__ __forceinline__ void gemm_section_128(
    const float* __restrict__ arow,   // this lane's A row, 128 floats
    float scale,                      // folded scatter-mean reciprocal
    const float* __restrict__ Wsec,   // weight section [128 x 128], row-major
    int hi, int mlo, v8f acc[8])
{
    for (int kb = 0; kb < DD; kb += 4) {
        const int k = kb + 2 * hi;
        const v2f av = *(const v2f*)(arow + k);
        v2f a;
        a.x = av.x * scale;
        a.y = av.y * scale;
#pragma unroll
        for (int t = 0; t < 8; ++t) {
            const float* wp = Wsec + (size_t)k * DD + t * 16 + mlo;
            v2f b;
            b.x = wp[0];
            b.y = wp[DD];
            acc[t] = __builtin_amdgcn_wmma_f32_16x16x4_f32(
                false, a, false, b, (short)0, acc[t], false, false);
        }
    }
}

// ---------------------------------------------------------------------------
// Fused: scatter-mean finalize + concat + MLP (fp32 WMMA) + layernorm.
// One wave computes one 16-row x 128-col output tile.
// ---------------------------------------------------------------------------
__global__ __launch_bounds__(WPB * 32) void mlp_ln_kernel(
    const float* __restrict__ node_attr,
    const float* __restrict__ sums_mesh, const float* __restrict__ cnt_mesh,
    const float* __restrict__ sums_con,  const float* __restrict__ cnt_con,
    const float* __restrict__ W1, const float* __restrict__ b1,
    const float* __restrict__ W2, const float* __restrict__ b2,
    const float* __restrict__ gamma, const float* __restrict__ beta,
    float* __restrict__ out, int n_nodes, int n_tiles)
{
    __shared__ float lds_h[WPB][TILE_M * HSTRIDE];

    const int wave = threadIdx.x >> 5;
    const int lane = threadIdx.x & 31;
    const int tile = blockIdx.x * WPB + wave;
    if (tile >= n_tiles) return;          // wave-uniform; EXEC stays all-1s

    const int mlo = lane & 15;            // A row / B col within tile
    const int hi  = lane >> 4;            // K sub-group
    int row = tile * TILE_M + mlo;
    if (row >= n_nodes) row = n_nodes - 1;   // clamp reads (tail)

    const float cm = cnt_mesh[row];
    const float cc = cnt_con[row];
    const float rmesh = 1.0f / (cm > 1.0f ? cm : 1.0f);
    const float rcon  = 1.0f / (cc > 1.0f ? cc : 1.0f);

    // ---------------- GEMM1: [16 x 384] @ W1[384 x 128] -------------------
    // Split into three branch-free 128-K sections (node | mesh | contact).
    v8f acc[8];
#pragma unroll
    for (int t = 0; t < 8; ++t) acc[t] = (v8f)0.0f;

    gemm_section_128(node_attr + (size_t)row * DD, 1.0f,
                     W1,                 hi, mlo, acc);
    gemm_section_128(sums_mesh + (size_t)row * DD, rmesh,
                     W1 + (size_t)DD * DD,     hi, mlo, acc);
    gemm_section_128(sums_con  + (size_t)row * DD, rcon,
                     W1 + (size_t)2 * DD * DD, hi, mlo, acc);

    // bias + relu, stage H tile into LDS to re-fragment for GEMM2
    float* __restrict__ hl = &lds_h[wave][0];
#pragma unroll
    for (int t = 0; t < 8; ++t) {
        const float bb = b1[t * 16 + mlo];
#pragma unroll
        for (int r = 0; r < 8; ++r) {
            float v = acc[t][r] + bb;
            v = v > 0.0f ? v : 0.0f;
            hl[(r + 8 * hi) * HSTRIDE + t * 16 + mlo] = v;
        }
    }

    // ---------------- GEMM2: H[16 x 128] @ W2[128 x 128] ------------------
    v8f acc2[8];
#pragma unroll
    for (int t = 0; t < 8; ++t) acc2[t] = (v8f)0.0f;

    {
        const float* __restrict__ ha = hl + mlo * HSTRIDE;
        for (int kb = 0; kb < DD; kb += 4) {
            const int k = kb + 2 * hi;
            v2f a;
            a.x = ha[k];
            a.y = ha[k + 1];
#pragma unroll
            for (int t = 0; t < 8; ++t) {
                const float* wp = W2 + (size_t)k * DD + t * 16 + mlo;
                v2f b;
                b.x = wp[0];
                b.y = wp[DD];
                acc2[t] = __builtin_amdgcn_wmma_f32_16x16x4_f32(
                    false, a, false, b, (short)0, acc2[t], false, false);
            }
        }
    }

    // bias, write H2 back into the same LDS slice (wave-local, DS in-order)
#pragma unroll
    for (int t = 0; t < 8; ++t) {
        const float bb = b2[t * 16 + mlo];
#pragma unroll
        for (int r = 0; r < 8; ++r) {
            hl[(r + 8 * hi) * HSTRIDE + t * 16 + mlo] = acc2[t][r] + bb;
        }
    }

    // ---------------- LayerNorm over last dim (128) -----------------------
    // Lane handles row=mlo, columns [hi*64, hi*64+64); partner lane^16 has
    // the other half -> combine with shfl_xor.
    const float* __restrict__ hr = hl + mlo * HSTRIDE + hi * 64;
    float s = 0.0f, ss = 0.0f;
#pragma unroll
    for (int c = 0; c < 64; ++c) {
        const float v = hr[c];
        s += v;
        ss += v * v;
    }
    s  += __shfl_xor(s, 16, 32);
    ss += __shfl_xor(ss, 16, 32);
    const float mean = s * (1.0f / 128.0f);
    const float var  = ss * (1.0f / 128.0f) - mean * mean;
    const float rstd = rsqrtf(var + 1e-5f);

    const int orow_idx = tile * TILE_M + mlo;
    if (orow_idx < n_nodes) {
        float* __restrict__ orow = out + (size_t)orow_idx * DD + hi * 64;
        const float* __restrict__ g  = gamma + hi * 64;
        const float* __restrict__ bt = beta  + hi * 64;
#pragma unroll
        for (int c = 0; c < 64; c += 4) {
            v4f hv = *(const v4f*)(hr + c);
            v4f gv = *(const v4f*)(g + c);
            v4f bv = *(const v4f*)(bt + c);
            v4f o;
            o.x = (hv.x - mean) * rstd * gv.x + bv.x;
            o.y = (hv.y - mean) * rstd * gv.y + bv.y;
            o.z = (hv.z - mean) * rstd * gv.z + bv.z;
            o.w = (hv.w - mean) * rstd * gv.w + bv.w;
            __builtin_nontemporal_store(o, (v4f*)(orow + c));
        }
    }
}

// ---------------------------------------------------------------------------
extern "C" void kernel_launch(void* const* d_in, const int* in_sizes, int n_in,
                              void* d_out, int out_size, void* d_ws, size_t ws_size,
                              hipStream_t stream) {
    const float* node_attr = (const float*)d_in[0];
    const float* edge_attr = (const float*)d_in[1];
    const float* edge_con  = (const float*)d_in[2];
    const int*   recv      = (const int*)d_in[3];
    const int*   recv_c    = (const int*)d_in[4];
    const float* W1        = (const float*)d_in[5];
    const float* b1        = (const float*)d_in[6];
    const float* W2        = (const float*)d_in[7];
    const float* b2        = (const float*)d_in[8];
    const float* ln_gamma  = (const float*)d_in[9];
    const float* ln_beta   = (const float*)d_in[10];
    float* out = (float*)d_out;

    const int n_nodes  = in_sizes[0] / DD;
    const int n_edges  = in_sizes[1] / DD;
    const int n_cedges = in_sizes[2] / DD;

    // workspace layout (floats): sums_mesh[N*128] | sums_con[N*128] | cnt_mesh[N] | cnt_con[N]
    float* sums_mesh = (float*)d_ws;
    float* sums_con  = sums_mesh + (size_t)n_nodes * DD;
    float* cnt_mesh  = sums_con  + (size_t)n_nodes * DD;
    float* cnt_con   = cnt_mesh  + (size_t)n_nodes;
    const size_t ws_floats = (size_t)n_nodes * (2 * DD + 2);

    hipMemsetAsync(d_ws, 0, ws_floats * sizeof(float), stream);

    scatter_sum_kernel<<<(n_edges + 7) / 8, 256, 0, stream>>>(
        edge_attr, recv, sums_mesh, cnt_mesh, n_edges);
    scatter_sum_kernel<<<(n_cedges + 7) / 8, 256, 0, stream>>>(
        edge_con, recv_c, sums_con, cnt_con, n_cedges);

    const int n_tiles = (n_nodes + TILE_M - 1) / TILE_M;
    mlp_ln_kernel<<<(n_tiles + WPB - 1) / WPB, WPB * 32, 0, stream>>>(
        node_attr, sums_mesh, cnt_mesh, sums_con, cnt_con,
        W1, b1, W2, b2, ln_gamma, ln_beta, out, n_nodes, n_tiles);
}